// NT_Xent_21320217657659
// MI455X (gfx1250) — compile-verified
//
#include <hip/hip_runtime.h>
#include <math.h>

#define B_DIM 4096
#define D_DIM 128
#define KC    32          // K-chunk staged in LDS
#define LDP   36          // padded LDS row stride (32 + 4 floats) -> conflict-free

typedef float v2f __attribute__((ext_vector_type(2)));
typedef float v8f __attribute__((ext_vector_type(8)));

// ---------------------------------------------------------------------------
// 1. Row L2 norms: grid 2*B, 128 threads/block. rows [0,B) -> z_i, [B,2B) -> z_j
// ---------------------------------------------------------------------------
__global__ __launch_bounds__(128) void rownorms(const float* __restrict__ zi,
                                                const float* __restrict__ zj,
                                                float* __restrict__ nI,
                                                float* __restrict__ nJ) {
    __shared__ float red[128];
    const int row = blockIdx.x;
    const float* z = (row < B_DIM) ? zi : zj;
    const int r = row & (B_DIM - 1);
    const float v = z[(size_t)r * D_DIM + threadIdx.x];
    red[threadIdx.x] = v * v;
    __syncthreads();
    for (int s = 64; s > 0; s >>= 1) {
        if (threadIdx.x < s) red[threadIdx.x] += red[threadIdx.x + s];
        __syncthreads();
    }
    if (threadIdx.x == 0) ((row < B_DIM) ? nI : nJ)[r] = sqrtf(red[0]);
}

// ---------------------------------------------------------------------------
// 2. Cosine-similarity GEMM: S = (A @ B^T) / (max(nA*nB, eps) * TEMP)
//    128x128 tile per 256-thread block; f32 WMMA 16x16x4 (full f32 precision).
//    8 waves: wave (wr in 0..3, wc in 0..1) owns a 32x64 sub-tile = 2x4 tiles.
//    Global->LDS staging uses CDNA5 async-LDS loads with a 2-deep double
//    buffer: next K-chunk streams in (ASYNCcnt) while current chunk's WMMAs run.
// ---------------------------------------------------------------------------
__global__ __launch_bounds__(256) void cosgemm(const float* __restrict__ A,
                                               const float* __restrict__ Bm,
                                               const float* __restrict__ nA,
                                               const float* __restrict__ nB,
                                               float* __restrict__ S) {
    __shared__ __align__(16) float lA[2][128 * LDP];
    __shared__ __align__(16) float lB[2][128 * LDP];

    const int tid  = threadIdx.x;
    const int lane = tid & 31;
    const int wave = tid >> 5;
    const int wr   = wave >> 1;   // 0..3 : 32-row band
    const int wc   = wave & 1;    // 0..1 : 64-col band
    const int rowBase = blockIdx.y * 128;
    const int colBase = blockIdx.x * 128;

    v8f acc[2][4];
    v8f cz = {};
#pragma unroll
    for (int tm = 0; tm < 2; ++tm)
#pragma unroll
        for (int tn = 0; tn < 4; ++tn) acc[tm][tn] = cz;

    // Cooperative loader mapping: thread -> (row 0..31, 4-float column group)
    const int ldRow = tid >> 3;        // 0..31
    const int ldCol = (tid & 7) * 4;   // 0,4,...,28 (16B groups)

    // Per-thread 32-bit global byte offsets (max ~2MB, fits u32).
    const unsigned gA0 = ((unsigned)(rowBase + ldRow) * D_DIM + ldCol) * 4u;
    const unsigned gB0 = ((unsigned)(colBase + ldRow) * D_DIM + ldCol) * 4u;

    // Issue one K-chunk's async global->LDS loads (8 instructions per wave).
    auto issue_chunk = [&](int buf, int kc) {
#pragma unroll
        for (int rr = 0; rr < 128; rr += 32) {
            const unsigned la = (unsigned)(uintptr_t)&lA[buf][(ldRow + rr) * LDP + ldCol];
            const unsigned lb = (unsigned)(uintptr_t)&lB[buf][(ldRow + rr) * LDP + ldCol];
            const unsigned ga = gA0 + (unsigned)(rr * D_DIM + kc) * 4u;
            const unsigned gb = gB0 + (unsigned)(rr * D_DIM + kc) * 4u;
            asm volatile("global_load_async_to_lds_b128 %0, %1, %2 offset:0"
                         :: "v"(la), "v"(ga), "s"(A) : "memory");
            asm volatile("global_load_async_to_lds_b128 %0, %1, %2 offset:0"
                         :: "v"(lb), "v"(gb), "s"(Bm) : "memory");
        }
    };

    issue_chunk(0, 0);   // prologue: chunk 0 -> buffer 0

    for (int c = 0; c < D_DIM / KC; ++c) {
        const int buf = c & 1;
        if (c < D_DIM / KC - 1) {
            issue_chunk(buf ^ 1, (c + 1) * KC);      // prefetch next chunk
            // In-order completion: <=8 outstanding means chunk c has landed.
            asm volatile("s_wait_asynccnt 8" ::: "memory");
        } else {
            asm volatile("s_wait_asynccnt 0" ::: "memory");
        }
        __syncthreads();   // all waves' portions of chunk c visible in LDS

#pragma unroll
        for (int k = 0; k < KC; k += 4) {
            // A 16x4 f32 frag: lanes 0-15 hold K=k,k+1 ; lanes 16-31 K=k+2,k+3
            const int kOff = k + ((lane >> 4) << 1);
            v2f aF[2], bF[4];
#pragma unroll
            for (int tm = 0; tm < 2; ++tm) {
                const int m = wr * 32 + tm * 16 + (lane & 15);
                aF[tm] = *(const v2f*)(&lA[buf][m * LDP + kOff]);
            }
#pragma unroll
            for (int tn = 0; tn < 4; ++tn) {
                const int n = wc * 64 + tn * 16 + (lane & 15);
                bF[tn] = *(const v2f*)(&lB[buf][n * LDP + kOff]);
            }
#pragma unroll
            for (int tm = 0; tm < 2; ++tm)
#pragma unroll
                for (int tn = 0; tn < 4; ++tn)
                    acc[tm][tn] = __builtin_amdgcn_wmma_f32_16x16x4_f32(
                        false, aF[tm], false, bF[tn], (short)0, acc[tm][tn],
                        false, false);
        }
        __syncthreads();   // done reading buf before it is refilled (c+2)
    }

    // Epilogue: C/D layout — VGPR v: lanes 0-15 -> M=v, lanes 16-31 -> M=v+8.
    float nbv[4];
#pragma unroll
    for (int tn = 0; tn < 4; ++tn)
        nbv[tn] = nB[colBase + wc * 64 + tn * 16 + (lane & 15)];
    const int mHalf = (lane >> 4) * 8;
#pragma unroll
    for (int tm = 0; tm < 2; ++tm) {
#pragma unroll
        for (int v = 0; v < 8; ++v) {
            const int m = rowBase + wr * 32 + tm * 16 + mHalf + v;
            const float nav = nA[m];
#pragma unroll
            for (int tn = 0; tn < 4; ++tn) {
                const int n = colBase + wc * 64 + tn * 16 + (lane & 15);
                const float denom = fmaxf(nav * nbv[tn], 1e-8f) * 0.5f; // * TEMP
                S[(size_t)m * B_DIM + n] = acc[tm][tn][v] / denom;
            }
        }
    }
}

// ---------------------------------------------------------------------------
// 3a. CE over a direct row r of S: -logp = lse(row) - row[(r+1) % B]
// ---------------------------------------------------------------------------
__global__ __launch_bounds__(256) void ce_direct(const float* __restrict__ S,
                                                 float* __restrict__ rl,
                                                 int slotBase) {
    __shared__ float red[256];
    const int r = blockIdx.x;
    const int tid = threadIdx.x;
    const float* row = S + (size_t)r * B_DIM;

    float v[16];
    float mx = -3.4e38f;
#pragma unroll
    for (int i = 0; i < 16; ++i) {
        v[i] = row[tid + i * 256];
        mx = fmaxf(mx, v[i]);
    }
    red[tid] = mx;
    __syncthreads();
    for (int s = 128; s > 0; s >>= 1) {
        if (tid < s) red[tid] = fmaxf(red[tid], red[tid + s]);
        __syncthreads();
    }
    mx = red[0];
    __syncthreads();
    float sum = 0.f;
#pragma unroll
    for (int i = 0; i < 16; ++i) sum += expf(v[i] - mx);
    red[tid] = sum;
    __syncthreads();
    for (int s = 128; s > 0; s >>= 1) {
        if (tid < s) red[tid] += red[tid + s];
        __syncthreads();
    }
    if (tid == 0) {
        const float lse = mx + logf(red[0]);
        const int lab = (r + 1) & (B_DIM - 1);
        rl[slotBase + r] = lse - row[lab];
    }
}

// ---------------------------------------------------------------------------
// 3b. CE over masked row m (diagonal dropped row-major, reshaped (B-1, B)):
//     flat off-diag index t -> orig (i = t/(B-1), j = j' + (j' >= i)), label 0.
// ---------------------------------------------------------------------------
__global__ __launch_bounds__(256) void ce_masked(const float* __restrict__ S,
                                                 float* __restrict__ rl,
                                                 int slotBase) {
    __shared__ float red[256];
    const int m = blockIdx.x;        // 0 .. B-2
    const int tid = threadIdx.x;

    float v[16];
    float mx = -3.4e38f;
#pragma unroll
    for (int i = 0; i < 16; ++i) {
        const int t  = m * B_DIM + tid + i * 256;
        const int ri = t / (B_DIM - 1);
        const int jp = t - ri * (B_DIM - 1);
        const int j  = jp + (jp >= ri ? 1 : 0);
        v[i] = S[(size_t)ri * B_DIM + j];
        mx = fmaxf(mx, v[i]);
    }
    red[tid] = mx;
    __syncthreads();
    for (int s = 128; s > 0; s >>= 1) {
        if (tid < s) red[tid] = fmaxf(red[tid], red[tid + s]);
        __syncthreads();
    }
    mx = red[0];
    __syncthreads();
    float sum = 0.f;
#pragma unroll
    for (int i = 0; i < 16; ++i) sum += expf(v[i] - mx);
    red[tid] = sum;
    __syncthreads();
    for (int s = 128; s > 0; s >>= 1) {
        if (tid < s) red[tid] += red[tid + s];
        __syncthreads();
    }
    if (tid == 0) {
        const float lse = mx + logf(red[0]);
        const int t0 = m * B_DIM;                 // label column 0 of masked row
        const int ri = t0 / (B_DIM - 1);
        const int jp = t0 - ri * (B_DIM - 1);
        const int j  = jp + (jp >= ri ? 1 : 0);
        rl[slotBase + m] = lse - S[(size_t)ri * B_DIM + j];
    }
}

// ---------------------------------------------------------------------------
// 4. Deterministic final reduction: loss = sum(rowloss) / (2*(2B-1))
// ---------------------------------------------------------------------------
__global__ __launch_bounds__(256) void finalize(const float* __restrict__ rl,
                                                float* __restrict__ out) {
    __shared__ float red[256];
    const int N = 2 * (2 * B_DIM - 1);
    float s = 0.f;
    for (int i = threadIdx.x; i < N; i += 256) s += rl[i];
    red[threadIdx.x] = s;
    __syncthreads();
    for (int st = 128; st > 0; st >>= 1) {
        if (threadIdx.x < st) red[threadIdx.x] += red[threadIdx.x + st];
        __syncthreads();
    }
    if (threadIdx.x == 0) out[0] = red[0] / (float)N;
}

// ---------------------------------------------------------------------------
extern "C" void kernel_launch(void* const* d_in, const int* in_sizes, int n_in,
                              void* d_out, int out_size, void* d_ws, size_t ws_size,
                              hipStream_t stream) {
    const float* z_i = (const float*)d_in[0];
    const float* z_j = (const float*)d_in[1];
    float* out = (float*)d_out;

    float* ws = (float*)d_ws;
    const size_t MAT = (size_t)B_DIM * B_DIM;
    float* Sij = ws;
    float* Sji = ws + MAT;
    float* Sii = ws + 2 * MAT;
    float* Sjj = ws + 3 * MAT;
    float* nI  = ws + 4 * MAT;
    float* nJ  = nI + B_DIM;
    float* rl  = nJ + B_DIM;   // 2*(2B-1) = 16382 row losses

    rownorms<<<2 * B_DIM, 128, 0, stream>>>(z_i, z_j, nI, nJ);

    dim3 g(B_DIM / 128, B_DIM / 128);
    cosgemm<<<g, 256, 0, stream>>>(z_i, z_j, nI, nJ, Sij);
    cosgemm<<<g, 256, 0, stream>>>(z_j, z_i, nJ, nI, Sji);
    cosgemm<<<g, 256, 0, stream>>>(z_i, z_i, nI, nI, Sii);
    cosgemm<<<g, 256, 0, stream>>>(z_j, z_j, nJ, nJ, Sjj);

    // loss_a slots [0, 2B-1), loss_b slots [2B-1, 2*(2B-1))
    ce_direct<<<B_DIM,     256, 0, stream>>>(Sij, rl, 0);
    ce_masked<<<B_DIM - 1, 256, 0, stream>>>(Sii, rl, B_DIM);
    ce_direct<<<B_DIM,     256, 0, stream>>>(Sji, rl, 2 * B_DIM - 1);
    ce_masked<<<B_DIM - 1, 256, 0, stream>>>(Sjj, rl, 3 * B_DIM - 1);

    finalize<<<1, 256, 0, stream>>>(rl, out);
}